// KHopGraphConvolution_72868415143955
// MI455X (gfx1250) — compile-verified
//
#include <hip/hip_runtime.h>

typedef __attribute__((ext_vector_type(2))) float v2f;
typedef __attribute__((ext_vector_type(8))) float v8f;

#define D_FEAT 256
#define LDS_STRIDE 260  // 256 + 4 dword pad to avoid LDS bank conflicts

// ---------------------------------------------------------------------------
// Zero-fill (graph-capture-safe scratch init)
// ---------------------------------------------------------------------------
__global__ void zero_kernel(float4* __restrict__ p, long n4) {
    long i = (long)blockIdx.x * blockDim.x + threadIdx.x;
    if (i < n4) p[i] = make_float4(0.f, 0.f, 0.f, 0.f);
}

// ---------------------------------------------------------------------------
// SPMM scatter: h_out[dst[e]] += val[e] * h_in[src[e]]
// One wave32 per edge; each lane owns 8 contiguous floats (two float4 gathers,
// fully coalesced 1KB per wave) and issues 8 native f32 atomic adds.
// ---------------------------------------------------------------------------
__global__ void spmm_scatter(const float* __restrict__ h_in,
                             const int*   __restrict__ esrc,
                             const int*   __restrict__ edst,
                             const float* __restrict__ eval,
                             float*       __restrict__ h_out,
                             int n_edges) {
    int wave = (int)((blockIdx.x * (long)blockDim.x + threadIdx.x) >> 5);
    int lane = threadIdx.x & 31;
    if (wave >= n_edges) return;

    int   s = esrc[wave];
    int   d = edst[wave];
    float v = eval[wave];

    const float4* in = (const float4*)(h_in + (size_t)s * D_FEAT);
    float*       out = h_out + (size_t)d * D_FEAT + lane * 8;

    float4 p0 = in[lane * 2 + 0];
    float4 p1 = in[lane * 2 + 1];

    unsafeAtomicAdd(out + 0, v * p0.x);
    unsafeAtomicAdd(out + 1, v * p0.y);
    unsafeAtomicAdd(out + 2, v * p0.z);
    unsafeAtomicAdd(out + 3, v * p0.w);
    unsafeAtomicAdd(out + 4, v * p1.x);
    unsafeAtomicAdd(out + 5, v * p1.y);
    unsafeAtomicAdd(out + 6, v * p1.z);
    unsafeAtomicAdd(out + 7, v * p1.w);
}

// ---------------------------------------------------------------------------
// Dense GEMM via V_WMMA_F32_16X16X4_F32 (full fp32 precision).
//   out[row0:row0+16, :] (+)= A[row0:row0+16, 0:256] @ B[0:256, 0:256] (+ bias)
// Block = 256 threads = 8 waves; block owns a 16-row stripe, each wave owns
// two 16x16 column tiles (cols wave*32 .. wave*32+31). A-stripe staged in LDS.
//
// ISA fragment layouts (CDNA5 ISA):
//   A 16x4 f32 : lanes 0-15 M=lane {K=k,k+1}; lanes 16-31 M=lane-16 {K=k+2,k+3}
//   B 4x16 f32 : lanes 0-15 N=lane {K=k,k+1}; lanes 16-31 N=lane-16 {K=k+2,k+3}
//   C/D 16x16  : n = lane&15, m = vgpr_r + 8*(lane>>4)
// ---------------------------------------------------------------------------
__global__ void gemm_wmma_f32(const float* __restrict__ A,
                              const float* __restrict__ B,
                              const float* __restrict__ bias,  // nullptr -> skip
                              float*       __restrict__ out,
                              int accumulate) {
    __shared__ float a_lds[16 * LDS_STRIDE];

    const int row0 = blockIdx.x * 16;
    const int tid  = threadIdx.x;

    // Cooperative coalesced load of the 16x256 A stripe (4 float4 per thread).
    for (int p = 0; p < 4; ++p) {
        int idx = p * 256 + tid;        // float4 slot 0..1023
        int r   = idx >> 6;             // 64 float4 per row
        int cc  = (idx & 63) * 4;
        float4 v = *(const float4*)(A + (size_t)(row0 + r) * D_FEAT + cc);
        float* q = &a_lds[r * LDS_STRIDE + cc];
        q[0] = v.x; q[1] = v.y; q[2] = v.z; q[3] = v.w;
    }
    __syncthreads();

    const int wave = tid >> 5;          // 0..7
    const int lane = tid & 31;
    const int half = lane >> 4;         // 0 or 1
    const int mn   = lane & 15;
    const int c0   = wave * 32;         // first of this wave's two col tiles

    v8f acc0 = {};
    v8f acc1 = {};
    if (accumulate) {
        for (int r = 0; r < 8; ++r) {
            size_t m = (size_t)(row0 + r + 8 * half) * D_FEAT;
            acc0[r] = out[m + c0 + mn];
            acc1[r] = out[m + c0 + 16 + mn];
        }
    }

    for (int k = 0; k < D_FEAT; k += 4) {
        // A fragment from LDS (8B-aligned ds_load_b64)
        v2f a = *(const v2f*)(&a_lds[mn * LDS_STRIDE + k + 2 * half]);

        // B fragments from global (W is 256KB -> cache resident)
        const int kb = k + 2 * half;
        const float* brow0 = B + (size_t)kb * D_FEAT;
        const float* brow1 = brow0 + D_FEAT;
        v2f b0, b1;
        b0.x = brow0[c0 + mn];       b0.y = brow1[c0 + mn];
        b1.x = brow0[c0 + 16 + mn];  b1.y = brow1[c0 + 16 + mn];

        acc0 = __builtin_amdgcn_wmma_f32_16x16x4_f32(
            false, a, false, b0, (short)0, acc0, false, false);
        acc1 = __builtin_amdgcn_wmma_f32_16x16x4_f32(
            false, a, false, b1, (short)0, acc1, false, false);
    }

    if (bias != nullptr) {
        float bv0 = bias[c0 + mn];
        float bv1 = bias[c0 + 16 + mn];
        for (int r = 0; r < 8; ++r) { acc0[r] += bv0; acc1[r] += bv1; }
    }

    for (int r = 0; r < 8; ++r) {
        size_t m = (size_t)(row0 + r + 8 * half) * D_FEAT;
        out[m + c0 + mn]      = acc0[r];
        out[m + c0 + 16 + mn] = acc1[r];
    }
}

// ---------------------------------------------------------------------------
// Launch: zero(h1); spmm(x->h1); out = h1@W0 + bias;
//         zero(h2); spmm(h1->h2); out += h2@W1
// ---------------------------------------------------------------------------
extern "C" void kernel_launch(void* const* d_in, const int* in_sizes, int n_in,
                              void* d_out, int out_size, void* d_ws, size_t ws_size,
                              hipStream_t stream) {
    const float* x    = (const float*)d_in[0];
    const int*   esrc = (const int*)d_in[1];
    const int*   edst = (const int*)d_in[2];
    const float* eval = (const float*)d_in[3];
    const float* W    = (const float*)d_in[4];   // [2, 256, 256]
    const float* bias = (const float*)d_in[5];   // [256]
    float* out = (float*)d_out;

    const int n_nodes = in_sizes[0] / D_FEAT;    // 50000
    const int n_edges = in_sizes[1];             // 1.6M

    const size_t feat_elems = (size_t)n_nodes * D_FEAT;
    float* h1 = (float*)d_ws;
    float* h2 = (float*)d_ws + feat_elems;

    const long n4 = (long)feat_elems / 4;
    const int  zblocks = (int)((n4 + 255) / 256);
    const int  eblocks = (n_edges + 7) / 8;      // 8 edges (waves) per block
    const int  gblocks = n_nodes / 16;           // 16-row stripes (50000 % 16 == 0)

    // Hop 1
    zero_kernel<<<zblocks, 256, 0, stream>>>((float4*)h1, n4);
    spmm_scatter<<<eblocks, 256, 0, stream>>>(x, esrc, edst, eval, h1, n_edges);
    gemm_wmma_f32<<<gblocks, 256, 0, stream>>>(h1, W, bias, out, 0);

    // Hop 2
    zero_kernel<<<zblocks, 256, 0, stream>>>((float4*)h2, n4);
    spmm_scatter<<<eblocks, 256, 0, stream>>>(h1, esrc, edst, eval, h2, n_edges);
    gemm_wmma_f32<<<gblocks, 256, 0, stream>>>(h2, W + D_FEAT * D_FEAT, nullptr, out, 1);
}